// HardwareConstrainedQuantizer_18502719111774
// MI455X (gfx1250) — compile-verified
//
#include <hip/hip_runtime.h>
#include <hip/hip_bf16.h>
#include <stdint.h>

// Streaming quantizer: out[i] = (w >  0.5f) ? 0.125f
//                               : (w <= -0.5f) ? -0.125f : 0.0f
// Memory-bound: ~268 MB total traffic -> ~11.5 us at 23.3 TB/s.
// Uses gfx1250 async global->LDS loads (ASYNCcnt) with wave-private
// double buffering + non-temporal B128 stores.

typedef float v4f __attribute__((ext_vector_type(4)));
typedef int   v4i __attribute__((vector_size(16)));   // matches builtin's V4i

#define BLOCK 256
#define TILE_FLOATS (BLOCK * 4)   // 1024 floats per block-tile (B128 per lane)
#define MAX_GRID 8192

#ifndef __has_builtin
#define __has_builtin(x) 0
#endif

#if defined(__HIP_DEVICE_COMPILE__) && defined(__gfx1250__) && \
    __has_builtin(__builtin_amdgcn_global_load_async_to_lds_b128)
#define HAVE_ASYNC_LDS 1
#else
#define HAVE_ASYNC_LDS 0
#endif

#if HAVE_ASYNC_LDS
#if __has_builtin(__builtin_amdgcn_s_wait_asynccnt)
#define WAIT_ASYNC_0() __builtin_amdgcn_s_wait_asynccnt(0)
#define WAIT_ASYNC_1() __builtin_amdgcn_s_wait_asynccnt(1)
#else
#define WAIT_ASYNC_0() asm volatile("s_wait_asynccnt 0" ::: "memory")
#define WAIT_ASYNC_1() asm volatile("s_wait_asynccnt 1" ::: "memory")
#endif

// Builtin expects (v4i addrspace(1)*, v4i addrspace(3)*, Ii, Ii).
// AS casts via integer round-trip: AS1 is the 64-bit flat address; AS3
// pointers are the low 32 bits of the generic address (LDS aperture layout).
typedef __attribute__((address_space(1))) v4i* as1_v4i_p;
typedef __attribute__((address_space(3))) v4i* as3_v4i_p;
#define AS1_PTR(p) ((as1_v4i_p)(uintptr_t)(p))
#define AS3_PTR(p) ((as3_v4i_p)(unsigned)(uintptr_t)(p))
#endif

__device__ __forceinline__ float quant1(float w) {
  // Exact semantics of the reference incl. argmin first-min tie-breaks:
  // w <= -0.5 -> -0.125 ; w > 0.5 -> +0.125 ; else 0 (covers |w|<1e-8 too).
  return (w > 0.5f) ? 0.125f : ((w <= -0.5f) ? -0.125f : 0.0f);
}

__device__ __forceinline__ v4f quant4(v4f v) {
  v4f q;
  q.x = quant1(v.x);
  q.y = quant1(v.y);
  q.z = quant1(v.z);
  q.w = quant1(v.w);
  return q;
}

__global__ __launch_bounds__(BLOCK) void hcq_kernel(const float* __restrict__ in,
                                                    float* __restrict__ out,
                                                    int fullTiles, long long n) {
  const int tid = (int)threadIdx.x;

#if HAVE_ASYNC_LDS
  // Wave-private staging: each lane async-copies its own 16B and reads back
  // the same 16B, so per-wave ASYNCcnt ordering suffices (no barriers).
  __shared__ v4f lds[2][BLOCK];
  const int stride = (int)gridDim.x;
  int t = (int)blockIdx.x;

  if (t < fullTiles) {
    __builtin_amdgcn_global_load_async_to_lds_b128(
        AS1_PTR(in + (size_t)t * TILE_FLOATS + tid * 4),
        AS3_PTR(&lds[0][tid]), 0, 0);
  }
  if (t + stride < fullTiles) {
    __builtin_amdgcn_global_load_async_to_lds_b128(
        AS1_PTR(in + (size_t)(t + stride) * TILE_FLOATS + tid * 4),
        AS3_PTR(&lds[1][tid]), 0, 0);
  }

  int buf = 0;
  while (t < fullTiles) {
    const int tNext = t + stride;
    if (tNext < fullTiles) {
      WAIT_ASYNC_1();   // one younger load outstanding; oldest is done
    } else {
      WAIT_ASYNC_0();   // nothing younger; drain fully
    }
    asm volatile("" ::: "memory");  // don't hoist the LDS read above the wait

    v4f v = lds[buf][tid];
    v4f q = quant4(v);
    __builtin_nontemporal_store(
        q, (v4f*)(out + (size_t)t * TILE_FLOATS + tid * 4));

    const int tPre = tNext + stride;
    if (tPre < fullTiles) {
      __builtin_amdgcn_global_load_async_to_lds_b128(
          AS1_PTR(in + (size_t)tPre * TILE_FLOATS + tid * 4),
          AS3_PTR(&lds[buf][tid]), 0, 0);
    }
    t = tNext;
    buf ^= 1;
  }
#else
  // Fallback: direct vectorized non-temporal streaming.
  const size_t nv4 = (size_t)fullTiles * BLOCK;
  for (size_t i = (size_t)blockIdx.x * BLOCK + tid; i < nv4;
       i += (size_t)gridDim.x * BLOCK) {
    v4f v = __builtin_nontemporal_load((const v4f*)in + i);
    __builtin_nontemporal_store(quant4(v), (v4f*)out + i);
  }
#endif

  // Scalar remainder (none for 4096x8192, but keep it correct in general).
  for (long long i = (long long)fullTiles * TILE_FLOATS +
                     (long long)blockIdx.x * BLOCK + tid;
       i < n; i += (long long)gridDim.x * BLOCK) {
    out[i] = quant1(in[i]);
  }
}

extern "C" void kernel_launch(void* const* d_in, const int* in_sizes, int n_in,
                              void* d_out, int out_size, void* d_ws, size_t ws_size,
                              hipStream_t stream) {
  (void)n_in; (void)d_ws; (void)ws_size; (void)out_size;
  const float* in = (const float*)d_in[0];
  float* out = (float*)d_out;
  const long long n = (long long)in_sizes[0];      // 4096*8192 = 33,554,432
  const int fullTiles = (int)(n / TILE_FLOATS);    // 32768 full tiles
  int grid = fullTiles < MAX_GRID ? (fullTiles > 0 ? fullTiles : 1) : MAX_GRID;
  hcq_kernel<<<grid, BLOCK, 0, stream>>>(in, out, fullTiles, n);
}